// deepLPM_5420248727600
// MI455X (gfx1250) — compile-verified
//
#include <hip/hip_runtime.h>
#include <math.h>

// Problem constants (fixed by the reference)
#define NPTS  300000
#define KCL   64
#define PD    32
#define NT    (NPTS / 16)        // 18750 16-row tiles, exact
#define WPB   4                  // waves per block
#define TPB   (WPB * 32)
#define DETC  1e-16f
#define PARTF (KCL*PD + 2*KCL)   // 2176 floats per partial: Mk | Nk | Sk
#define MAXBLK 384

typedef __attribute__((ext_vector_type(2))) float v2f;
typedef __attribute__((ext_vector_type(8))) float v8f;

__global__ __launch_bounds__(TPB) void deeplpm_em_kernel(
    const float* __restrict__ mu_phi,      // [N,32]
    const float* __restrict__ log_cov_phi, // [N]
    const float* __restrict__ pi_k,        // [64]
    const float* __restrict__ mu_k,        // [64,32]
    const float* __restrict__ log_cov_k,   // [64]
    float* __restrict__ gamma_out,         // [N,64]
    float* __restrict__ partials,          // [nblocks][2176]
    int niter, int totalWaves)
{
  __shared__ float muk_s[KCL * PD];             // 2048: mu_k staged for B-fragments
  __shared__ float acc_s[PARTF];                // 2176: block accumulators Mk|Nk|Sk
  __shared__ float wbuf[WPB][1584];             // per wave: phi 512 | gamma 1024 | t 48

  const int tid  = threadIdx.x;
  const int wid  = tid >> 5;
  const int lane = tid & 31;
  const int lo   = lane & 15;
  const int hi   = lane >> 4;

  // block-level staging + zeroing
  for (int e = tid; e < KCL * PD; e += TPB) muk_s[e] = mu_k[e];
  for (int e = tid; e < PARTF;    e += TPB) acc_s[e] = 0.0f;
  __syncthreads();

  float* phi_s = wbuf[wid];          // [16][32] row-major
  float* glds  = wbuf[wid] + 512;    // [16][64] row-major
  float* lcps  = wbuf[wid] + 1536;   // [16]
  float* elcps = wbuf[wid] + 1552;   // [16]
  float* a2s   = wbuf[wid] + 1568;   // [16]

  // per-lane, loop-invariant cluster constants (cluster k = c*16 + lo)
  float b2c[4], pic[4], lckc[4], invc[4];
  #pragma unroll
  for (int c = 0; c < 4; ++c) {
    int k = c * 16 + lo;
    float s = 0.0f;
    #pragma unroll
    for (int d = 0; d < PD; ++d) { float m = muk_s[k * PD + d]; s += m * m; }
    b2c[c]  = s;
    pic[c]  = pi_k[k];
    lckc[c] = log_cov_k[k];
    invc[c] = expf(-lckc[c]);
  }

  const v8f zero8 = {0.f,0.f,0.f,0.f,0.f,0.f,0.f,0.f};
  v8f macc[4][2];                    // gamma^T @ phi accumulators: [cluster grp][dim grp]
  #pragma unroll
  for (int c = 0; c < 4; ++c) { macc[c][0] = zero8; macc[c][1] = zero8; }
  float nk_loc[4] = {0.f,0.f,0.f,0.f};
  float sk_loc[4] = {0.f,0.f,0.f,0.f};

  const int gwave = blockIdx.x * WPB + wid;

  for (int iter = 0; iter < niter; ++iter) {
    int tile = gwave + iter * totalWaves;
    if (tile < NT) {
      // ---- stage 16x32 phi tile (contiguous 2KB) into LDS, coalesced ----
      const float4* src4 = (const float4*)(mu_phi + (size_t)tile * 512);
      float4*       dst4 = (float4*)phi_s;
      #pragma unroll
      for (int j = 0; j < 4; ++j) dst4[lane + j * 32] = src4[lane + j * 32];
      __builtin_amdgcn_wave_barrier();

      // ---- per-row stats: a2 = ||phi_row||^2, lcp, exp(lcp) ----
      float s = 0.0f;
      #pragma unroll
      for (int d = 0; d < 16; ++d) { float t = phi_s[lo * 32 + hi * 16 + d]; s += t * t; }
      s += __shfl_xor(s, 16, 32);
      if (hi == 0) {
        a2s[lo] = s;
        float lc = log_cov_phi[tile * 16 + lo];
        lcps[lo]  = lc;
        elcps[lo] = expf(lc);
      }
      __builtin_amdgcn_wave_barrier();

      // ---- GEMM1: dot = phi(16x32) @ mu_k^T -> 16x64 via f32 WMMA, K=4 x 8 ----
      v8f gam[4] = {zero8, zero8, zero8, zero8};
      #pragma unroll
      for (int kk = 0; kk < 8; ++kk) {
        int kb = kk * 4 + hi * 2;
        v2f a = *(const v2f*)&phi_s[lo * 32 + kb];          // A: row=lo, K pair
        #pragma unroll
        for (int c = 0; c < 4; ++c) {
          v2f b = *(const v2f*)&muk_s[(c * 16 + lo) * 32 + kb]; // B[k][n]=mu_k[n][k]
          gam[c] = __builtin_amdgcn_wmma_f32_16x16x4_f32(
              false, a, false, b, (short)0, gam[c], false, false);
        }
      }

      // ---- epilogue: KL -> w -> rowsum -> gamma ----
      float lcpr[8], elcpr[8], a2r[8];
      #pragma unroll
      for (int v = 0; v < 8; ++v) {
        int m = v + 8 * hi;
        lcpr[v] = lcps[m]; elcpr[v] = elcps[m]; a2r[v] = a2s[m];
      }
      #pragma unroll
      for (int c = 0; c < 4; ++c)
        #pragma unroll
        for (int v = 0; v < 8; ++v) {
          float dot = gam[c][v];
          float sq  = a2r[v] + b2c[c] - 2.0f * dot;
          float KL  = 0.5f * (32.0f * (lckc[c] - lcpr[v]) - 32.0f
                              + 32.0f * elcpr[v] * invc[c] + sq * invc[c]);
          gam[c][v] = pic[c] * expf(-KL);
        }
      #pragma unroll
      for (int v = 0; v < 8; ++v) {
        float rs = gam[0][v] + gam[1][v] + gam[2][v] + gam[3][v];
        rs += __shfl_xor(rs, 1, 16);
        rs += __shfl_xor(rs, 2, 16);
        rs += __shfl_xor(rs, 4, 16);
        rs += __shfl_xor(rs, 8, 16);
        #pragma unroll
        for (int c = 0; c < 4; ++c) gam[c][v] = gam[c][v] / rs + DETC;
      }

      // ---- N_k / S_k partials from fragment layout ----
      #pragma unroll
      for (int c = 0; c < 4; ++c)
        #pragma unroll
        for (int v = 0; v < 8; ++v) {
          float g = gam[c][v];
          nk_loc[c] += g;
          sk_loc[c] += g * (32.0f * elcpr[v] + a2r[v]);
        }

      // ---- stage gamma tile to LDS (for transpose-read + coalesced store) ----
      #pragma unroll
      for (int c = 0; c < 4; ++c)
        #pragma unroll
        for (int v = 0; v < 8; ++v)
          glds[(v + 8 * hi) * 64 + c * 16 + lo] = gam[c][v];
      __builtin_amdgcn_wave_barrier();

      // ---- coalesced gamma write: tile is a contiguous 4KB run of d_out ----
      float4* gout4 = (float4*)(gamma_out + (size_t)tile * 1024);
      const float4* gl4 = (const float4*)glds;
      #pragma unroll
      for (int j = 0; j < 8; ++j) gout4[lane + j * 32] = gl4[lane + j * 32];

      // ---- GEMM2: macc += gamma^T(64x16) @ phi(16x32), K=4 x 4 ----
      #pragma unroll
      for (int kk = 0; kk < 4; ++kk) {
        int kb = kk * 4 + hi * 2;
        v2f bfr[2];
        #pragma unroll
        for (int d = 0; d < 2; ++d) {
          bfr[d].x = phi_s[kb * 32 + d * 16 + lo];
          bfr[d].y = phi_s[(kb + 1) * 32 + d * 16 + lo];
        }
        #pragma unroll
        for (int c = 0; c < 4; ++c) {
          v2f af;
          af.x = glds[kb * 64 + c * 16 + lo];        // A[m=cluster][k=point]
          af.y = glds[(kb + 1) * 64 + c * 16 + lo];
          #pragma unroll
          for (int d = 0; d < 2; ++d)
            macc[c][d] = __builtin_amdgcn_wmma_f32_16x16x4_f32(
                false, af, false, bfr[d], (short)0, macc[c][d], false, false);
        }
      }
    }
  }

  // ---- block reduction: waves -> LDS accumulators (ds_add_f32) ----
  __syncthreads();
  float* Mk = acc_s;
  float* Nk = acc_s + KCL * PD;
  float* Sk = acc_s + KCL * PD + KCL;
  #pragma unroll
  for (int c = 0; c < 4; ++c) {
    #pragma unroll
    for (int d = 0; d < 2; ++d)
      #pragma unroll
      for (int v = 0; v < 8; ++v)
        atomicAdd(&Mk[(c * 16 + v + 8 * hi) * PD + d * 16 + lo], macc[c][d][v]);
    atomicAdd(&Nk[c * 16 + lo], nk_loc[c]);
    atomicAdd(&Sk[c * 16 + lo], sk_loc[c]);
  }
  __syncthreads();

  float* mypart = partials + (size_t)blockIdx.x * PARTF;
  for (int e = tid; e < PARTF; e += TPB) mypart[e] = acc_s[e];
}

__global__ __launch_bounds__(256) void deeplpm_finalize_kernel(
    const float* __restrict__ partials, int nparts,
    float* __restrict__ pi_out,       // [64]
    float* __restrict__ mu_out,       // [64,32]
    float* __restrict__ lcov_out)     // [64]
{
  __shared__ float fin[PARTF];
  const int tid = threadIdx.x;

  // deterministic fixed-order reduction over per-block partials
  for (int e = tid; e < PARTF; e += 256) {
    float s = 0.0f;
    for (int p = 0; p < nparts; ++p) s += partials[(size_t)p * PARTF + e];
    fin[e] = s;
  }
  __syncthreads();

  if (tid < KCL) pi_out[tid] = fin[KCL * PD + tid] * (1.0f / (float)NPTS);

  for (int e = tid; e < KCL * PD; e += 256) {
    int k = e >> 5;
    float mu = fin[e] / fin[KCL * PD + k];
    mu_out[e] = mu;
    fin[e] = mu;   // stash mu_new for ||mu||^2 (each e owned by one thread)
  }
  __syncthreads();

  if (tid < KCL) {
    float msq = 0.0f;
    #pragma unroll
    for (int d = 0; d < PD; ++d) { float m = fin[tid * PD + d]; msq += m * m; }
    float Nkv = fin[KCL * PD + tid];
    float Skv = fin[KCL * PD + KCL + tid];
    float cov = (Skv - Nkv * msq) / (32.0f * Nkv);
    lcov_out[tid] = logf(cov);
  }
}

extern "C" void kernel_launch(void* const* d_in, const int* in_sizes, int n_in,
                              void* d_out, int out_size, void* d_ws, size_t ws_size,
                              hipStream_t stream)
{
  (void)in_sizes; (void)n_in; (void)out_size;
  const float* mu_phi      = (const float*)d_in[0];
  const float* log_cov_phi = (const float*)d_in[1];
  const float* pi_k        = (const float*)d_in[2];
  const float* mu_k        = (const float*)d_in[3];
  const float* log_cov_k   = (const float*)d_in[4];
  // d_in[5] is P == 32 (compile-time constant here)

  float* out      = (float*)d_out;
  float* gamma    = out;                               // [N,64]
  float* pi_new   = out + (size_t)NPTS * KCL;          // [64]
  float* mu_new   = pi_new + KCL;                      // [64,32]
  float* lcov_new = mu_new + KCL * PD;                 // [64]

  int maxParts = (int)(ws_size / (PARTF * sizeof(float)));
  int nblocks  = maxParts < MAXBLK ? maxParts : MAXBLK;
  if (nblocks < 1) nblocks = 1;
  int totalWaves = nblocks * WPB;
  int niter = (NT + totalWaves - 1) / totalWaves;

  deeplpm_em_kernel<<<nblocks, TPB, 0, stream>>>(
      mu_phi, log_cov_phi, pi_k, mu_k, log_cov_k,
      gamma, (float*)d_ws, niter, totalWaves);

  deeplpm_finalize_kernel<<<1, 256, 0, stream>>>(
      (const float*)d_ws, nblocks, pi_new, mu_new, lcov_new);
}